// RGLRU_43963285242532
// MI455X (gfx1250) — compile-verified
//
#include <hip/hip_runtime.h>

// RG-LRU fused kernel for MI455X (gfx1250), B=4 L=2048 W=2048 H=16 D=128.
// One persistent block per (batch, head); gates via V_WMMA_F32_16X16X4_F32,
// x-tiles streamed with GLOBAL_LOAD_ASYNC_TO_LDS_B128 double-buffering,
// linear recurrence carried in registers with wave32 shuffles.

#define Bn 4
#define Ln 2048
#define Wn 2048
#define Hn 16
#define Dn 128
#define NT (Ln / 16)
#define LDS_STRIDE 132   // 128 + 4 pad -> conflict-free column access

typedef __attribute__((ext_vector_type(2))) float v2f;
typedef __attribute__((ext_vector_type(8))) float v8f;
typedef __attribute__((address_space(3))) float lds_f32;

__device__ __forceinline__ float sigmoidf_(float v) {
    return 1.0f / (1.0f + __expf(-v));
}

__global__ __launch_bounds__(256)
void rglru_fused_kernel(const float* __restrict__ x,
                        const float* __restrict__ rp,
                        const float* __restrict__ wx,
                        const float* __restrict__ bxv,
                        const float* __restrict__ wa,
                        const float* __restrict__ bav,
                        float* __restrict__ out) {
    __shared__ float xs[2][16 * LDS_STRIDE];

    const int b    = blockIdx.x / Hn;
    const int h    = blockIdx.x % Hn;
    const int tid  = threadIdx.x;
    const int lane = tid & 31;
    const int wv   = tid >> 5;        // wave id = 16-col n-tile within head
    const int hi   = lane >> 4;       // half-wave: 0 -> rows 0..7, 1 -> rows 8..15
    const int n    = lane & 15;       // N index within 16x16 tile
    const int col  = wv * 16 + n;     // channel within head [0,128)
    const int wglob = h * Dn + col;   // global channel [0,2048)

    const float* xb = x   + (size_t)b * Ln * Wn + h * Dn;
    float*       yb = out + (size_t)b * Ln * Wn + h * Dn;

    // Async copy of one 16x128 f32 tile into LDS buffer `buf`.
    // 512 chunks of 16B; each thread issues 2 GLOBAL_LOAD_ASYNC_TO_LDS_B128.
    auto issue_tile = [&](int buf, int t0) {
#pragma unroll
        for (int j = 0; j < 2; ++j) {
            const int q    = tid * 2 + j;
            const int rrow = q >> 5;          // 32 chunks per 128-float row
            const int cc   = (q & 31) * 4;
            const float* gp = xb + (size_t)(t0 + rrow) * Wn + cc;
            const unsigned int laddr =
                (unsigned int)(size_t)(lds_f32*)&xs[buf][rrow * LDS_STRIDE + cc];
            asm volatile("global_load_async_to_lds_b128 %0, %1, off"
                         :: "v"(laddr), "v"(gp) : "memory");
        }
    };

    // ---- Preload B-matrix (weight) fragments for both gates into VGPRs.
    // WMMA f32 16x16x4 B layout: lanes 0-15 hold K={k0,k0+1}, lanes 16-31 K={k0+2,k0+3}.
    float bWx0[32], bWx1[32], bWa0[32], bWa1[32];
#pragma unroll
    for (int ks = 0; ks < 32; ++ks) {
        const int kk   = 4 * ks + 2 * hi;
        const int base = (h * Dn + kk) * Dn + col;   // w[h, kk, col]
        bWx0[ks] = wx[base];       bWx1[ks] = wx[base + Dn];
        bWa0[ks] = wa[base];       bWa1[ks] = wa[base + Dn];
    }

    const float biasx = bxv[wglob];
    const float biasa = bav[wglob];
    const float rv    = rp[wglob];
    // log_a = -8 * gate_a * softplus(rp)  ->  coef = -8 * softplus(rp)
    const float coef  = -8.0f * __logf(1.0f + __expf(rv));

    issue_tile(0, 0);                 // prologue: tile 0 in flight
    float hcarry = 0.0f;              // recurrence state (h_{-1} = 0)

    for (int tt = 0; tt < NT; ++tt) {
        const int t0  = tt * 16;
        const int cur = tt & 1;

        if (tt + 1 < NT) {
            issue_tile(cur ^ 1, t0 + 16);                 // overlap next tile
            asm volatile("s_wait_asynccnt 0x2" ::: "memory");  // tile tt done
        } else {
            asm volatile("s_wait_asynccnt 0x0" ::: "memory");
        }
        __syncthreads();              // all waves' async writes visible

        if (tt + 2 < NT) {            // deep L2 prefetch, 2 tiles ahead
            __builtin_prefetch(xb + (size_t)(t0 + 32 + (tid >> 4)) * Wn + ((tid * 8) & 127), 0, 0);
        }

        const float* xt = xs[cur];

        // ---- Gate matmuls: D = A(16x128) x W(128x16), K-loop of 32 WMMA steps.
        v8f accX = {0.f, 0.f, 0.f, 0.f, 0.f, 0.f, 0.f, 0.f};
        v8f accA = {0.f, 0.f, 0.f, 0.f, 0.f, 0.f, 0.f, 0.f};
#pragma unroll
        for (int ks = 0; ks < 32; ++ks) {
            const int k = 4 * ks + 2 * hi;   // A layout mirrors B layout in K
            v2f af;
            af.x = xt[n * LDS_STRIDE + k];
            af.y = xt[n * LDS_STRIDE + k + 1];
            v2f bfx; bfx.x = bWx0[ks]; bfx.y = bWx1[ks];
            v2f bfa; bfa.x = bWa0[ks]; bfa.y = bWa1[ks];
            accX = __builtin_amdgcn_wmma_f32_16x16x4_f32(
                false, af, false, bfx, (short)0, accX, false, false);
            accA = __builtin_amdgcn_wmma_f32_16x16x4_f32(
                false, af, false, bfa, (short)0, accA, false, false);
        }

        // ---- Elementwise gates + local 8-step scan (this lane's rows).
        // C layout: vgpr i -> row (i + 8*hi), column n (same for both halves).
        float yv[8], Pv[8];
        float P = 1.0f, hl = 0.0f;
#pragma unroll
        for (int i = 0; i < 8; ++i) {
            const int m  = i + 8 * hi;
            const float gx = sigmoidf_(accX[i] + biasx);
            const float ga = sigmoidf_(accA[i] + biasa);
            const float a  = __expf(coef * ga);
            float mult = sqrtf(fmaxf(1.0f - a * a, 0.0f));
            if (t0 == 0 && m == 0) mult = 1.0f;     // first step unscaled
            const float nx = xt[m * LDS_STRIDE + col] * gx * mult;
            hl = fmaf(a, hl, nx);   // local scan, zero carry-in
            P *= a;                 // prefix product of a
            yv[i] = hl;
            Pv[i] = P;
        }

        // ---- Thread the carry: global carry -> rows 0..7 -> rows 8..15 -> next tile.
        // y_t = y_local_t + P_t * carry_in
        const float lowEnd = fmaf(Pv[7], hcarry, yv[7]);     // valid on lanes 0-15
        const float c2     = __shfl(lowEnd, n, 32);          // lower half's end
        const float myc    = hi ? c2 : hcarry;

        float lastv = 0.0f;
#pragma unroll
        for (int i = 0; i < 8; ++i) {
            const float yo = fmaf(Pv[i], myc, yv[i]);
            yb[(size_t)(t0 + i + 8 * hi) * Wn + col] = yo;
            lastv = yo;
        }
        hcarry = __shfl(lastv, n + 16, 32);   // upper half's end = h at tile end

        __syncthreads();   // everyone done with xs[cur] before it is re-filled
    }

    // Tail output: y[:, -1]  (one writer per channel)
    if (hi == 0) {
        out[(size_t)Bn * Ln * Wn + (size_t)b * Wn + wglob] = hcarry;
    }
}

extern "C" void kernel_launch(void* const* d_in, const int* in_sizes, int n_in,
                              void* d_out, int out_size, void* d_ws, size_t ws_size,
                              hipStream_t stream) {
    (void)in_sizes; (void)n_in; (void)out_size; (void)d_ws; (void)ws_size;
    const float* x  = (const float*)d_in[0];
    const float* rp = (const float*)d_in[1];
    const float* wx = (const float*)d_in[2];
    const float* bx = (const float*)d_in[3];
    const float* wa = (const float*)d_in[4];
    const float* ba = (const float*)d_in[5];
    float* out = (float*)d_out;

    rglru_fused_kernel<<<dim3(Bn * Hn), dim3(256), 0, stream>>>(
        x, rp, wx, bx, wa, ba, out);
}